// EncoderUniformSelection_7705171329186
// MI455X (gfx1250) — compile-verified
//
#include <hip/hip_runtime.h>
#include <hip/hip_bf16.h>

// ---------------------------------------------------------------------------
// EncoderUniformSelection for MI455X (gfx1250, wave32, WMMA).
// Roofline: reading x (134 MB fp32) dominates -> ~5.8us @ 23.3 TB/s.
// GEMM1 (131072x256x64) = 4.3 GFLOP fp32 on the matrix pipe via
// V_WMMA_F32_16X16X4_F32 (exact fp32, compute is free vs HBM).
// Round-2 change: W1 is staged in LDS K-pair-interleaved so every B
// fragment is a single 8B-aligned v2f LDS load (ds_load_b64) instead of
// two scalar b32 loads + v_mov shuffling -> cuts VALU/DS overhead per WMMA
// in the only hot loop.
// ---------------------------------------------------------------------------

typedef float v2f __attribute__((ext_vector_type(2)));
typedef float v8f __attribute__((ext_vector_type(8)));

__device__ __forceinline__ v8f wmma_f32x4(v2f a, v2f b, v8f c) {
    // 8-arg form: (neg_a, A, neg_b, B, c_mod, C, reuse_a, reuse_b)
    return __builtin_amdgcn_wmma_f32_16x16x4_f32(
        false, a, false, b, (short)0, c, false, false);
}

#define BN   32
#define TN   4096
#define DN   256
#define HN   64
#define AN   259
#define APAD 260
#define KD   256
#define DM   1024
#define NPX  16
#define STEP 273   // linspace(0,4095,16) step == 4095/15

#define SWST 160   // W1 LDS stride per k-pair row: 160 ≡ 32 (mod 64 banks)
#define SXST 68    // x LDS stride per row

// ---------------------------------------------------------------------------
// Kernel 1: sal[b,t] = sigmoid( relu(x@W1 + b1) @ W2 + b2 )
// 1024 blocks x 256 threads (8 waves). Each wave owns 16 rows x full N=64.
// ---------------------------------------------------------------------------
__global__ __launch_bounds__(256) void k_sal(
    const float* __restrict__ x,  const float* __restrict__ W1,
    const float* __restrict__ b1, const float* __restrict__ W2,
    const float* __restrict__ b2, float* __restrict__ sal)
{
    __shared__ float sW[32 * SWST];   // W1 K-chunk, pair-interleaved:
                                      // sW[(k>>1)*SWST + 2n + (k&1)]
    __shared__ float sX[128 * SXST];  // x tile: 128 rows x 64 k

    const int tid  = threadIdx.x;
    const int lane = tid & 31;
    const int w    = tid >> 5;        // wave 0..7
    const int l15  = lane & 15;
    const int hi   = lane >> 4;
    const long rowBase = (long)blockIdx.x * 128;

    v8f acc[4] = {};   // 16x64 output per wave (4 N-tiles of 16)

    for (int kc = 0; kc < DN; kc += 64) {
        // stage W1[kc..kc+63][0..63], pair-interleaved for v2f B-frag loads
        for (int i = tid; i < 64 * 64; i += 256) {
            int k = i >> 6, n = i & 63;
            sW[(k >> 1) * SWST + n * 2 + (k & 1)] = W1[(kc + k) * HN + n];
        }
        // stage x tile (float4 per thread, fully coalesced 16B global loads)
        for (int q = tid; q < 2048; q += 256) {
            int r = q >> 4, c4 = q & 15;
            const float4 v = *(const float4*)&x[(rowBase + r) * DN + kc + c4 * 4];
            *(float4*)&sX[r * SXST + c4 * 4] = v;
        }
        __syncthreads();

        const float* arow = &sX[(w * 16 + l15) * SXST + 2 * hi];
        const float* bcol = &sW[hi * SWST + 2 * l15];
        #pragma unroll 4
        for (int kk = 0; kk < 64; kk += 4) {
            const v2f a = *(const v2f*)&arow[kk];
            const float* brow = &bcol[(kk >> 1) * SWST];
            #pragma unroll
            for (int t = 0; t < 4; ++t) {
                const v2f bb = *(const v2f*)&brow[32 * t];   // n-tile t
                acc[t] = wmma_f32x4(a, bb, acc[t]);
            }
        }
        __syncthreads();
    }

    // Fused epilogue: relu(h + b1) . W2 + b2, sigmoid, lane-shuffle reduce.
    float b1c[4], w2c[4];
    #pragma unroll
    for (int t = 0; t < 4; ++t) {
        b1c[t] = b1[t * 16 + l15];
        w2c[t] = W2[t * 16 + l15];
    }
    const float b2v = b2[0];
    #pragma unroll
    for (int r = 0; r < 8; ++r) {
        float s = 0.f;
        #pragma unroll
        for (int t = 0; t < 4; ++t) {
            float hv = acc[t][r] + b1c[t];
            hv = hv > 0.f ? hv : 0.f;
            s += hv * w2c[t];
        }
        #pragma unroll
        for (int off = 8; off >= 1; off >>= 1)
            s += __shfl_xor(s, off, 32);     // reduce within each 16-lane half
        if (l15 == 0) {
            long row = rowBase + w * 16 + r + 8 * hi;  // C layout: M = r + 8*hi
            sal[row] = 1.0f / (1.0f + __expf(-(s + b2v)));
        }
    }
}

// ---------------------------------------------------------------------------
// Kernel 2: y_star pattern (1.0 exactly at t = 273*j)
// ---------------------------------------------------------------------------
__global__ __launch_bounds__(256) void k_ystar(float* __restrict__ ys)
{
    int i = blockIdx.x * 256 + threadIdx.x;
    if (i < BN * TN) {
        int t = i & (TN - 1);
        ys[i] = (t % STEP == 0) ? 1.0f : 0.0f;
    }
}

// ---------------------------------------------------------------------------
// Kernel 3: per-batch prefix sums of sal at the 16 cut points -> cum/T
// ---------------------------------------------------------------------------
__global__ __launch_bounds__(256) void k_cum(
    const float* __restrict__ sal, float* __restrict__ cumv)
{
    __shared__ float part[256];
    const int b = blockIdx.x, tid = threadIdx.x;
    const float* s = sal + b * TN;
    float acc = 0.f;
    #pragma unroll 4
    for (int i = 0; i < 16; ++i) acc += s[tid * 16 + i];
    part[tid] = acc;
    __syncthreads();
    if (tid < NPX) {
        int n = tid * STEP + 1;            // inclusive element count
        int comp = n >> 4, rem = n & 15;
        float c = 0.f;
        for (int i = 0; i < comp; ++i) c += part[i];
        for (int r = 0; r < rem; ++r) c += s[comp * 16 + r];
        cumv[b * NPX + tid] = c * (1.0f / (float)TN);
    }
}

// ---------------------------------------------------------------------------
// Kernel 4: build normalized/standardized anchor rows z (512 x 260, K-padded)
// ---------------------------------------------------------------------------
__global__ __launch_bounds__(256) void k_z(
    const float* __restrict__ x, const float* __restrict__ sal,
    const float* __restrict__ cumv, const float* __restrict__ mu,
    const float* __restrict__ sigma, float* __restrict__ zbuf)
{
    __shared__ float sa[APAD];
    __shared__ float red[256];
    const int blk = blockIdx.x;          // b*16 + j
    const int b = blk >> 4, j = blk & 15;
    const int t = j * STEP;
    const int tid = threadIdx.x;

    sa[tid] = x[((long)(b * TN + t)) * DN + tid];
    if (tid == 0) {
        sa[256] = sal[b * TN + t];
        sa[257] = (float)t / (float)TN;
        sa[258] = cumv[blk];
        sa[259] = 0.f;
    }
    __syncthreads();

    float v = sa[tid] * sa[tid];
    if (tid < 3) v += sa[256 + tid] * sa[256 + tid];
    red[tid] = v;
    __syncthreads();
    for (int s = 128; s > 0; s >>= 1) {
        if (tid < s) red[tid] += red[tid + s];
        __syncthreads();
    }
    const float inv = 1.0f / (sqrtf(red[0]) + 1e-6f);

    for (int i = tid; i < APAD; i += 256) {
        float zv = 0.f;
        if (i < AN) zv = (sa[i] * inv - mu[i]) / sigma[i];
        zbuf[blk * APAD + i] = zv;
    }
}

// ---------------------------------------------------------------------------
// Kernel 5: lifted = tanh(z @ Wl + bl).  512x260x256 fp32 WMMA, wave/tile.
// ---------------------------------------------------------------------------
__global__ __launch_bounds__(32) void k_lift(
    const float* __restrict__ zbuf, const float* __restrict__ Wl,
    const float* __restrict__ bl, float* __restrict__ lifted)
{
    const int m0 = blockIdx.x * 16, n0 = blockIdx.y * 16;
    const int lane = threadIdx.x & 31, l15 = lane & 15, hi = lane >> 4;
    v8f acc = {};
    const float* zr = zbuf + (m0 + l15) * APAD;
    for (int k = 0; k < APAD; k += 4) {
        const int ka = k + 2 * hi;
        v2f a;  a.x = zr[ka];  a.y = zr[ka + 1];
        v2f bb;
        bb.x = (ka     < AN) ? Wl[ka       * KD + n0 + l15] : 0.f;
        bb.y = (ka + 1 < AN) ? Wl[(ka + 1) * KD + n0 + l15] : 0.f;
        acc = wmma_f32x4(a, bb, acc);
    }
    const float blv = bl[n0 + l15];
    #pragma unroll
    for (int r = 0; r < 8; ++r) {
        int m = m0 + r + 8 * hi;
        lifted[m * KD + n0 + l15] = tanhf(acc[r] + blv);
    }
}

// ---------------------------------------------------------------------------
// Kernel 6: tokens = lifted @ Wp + bp.  512x256x1024 fp32 WMMA, wave/tile.
// ---------------------------------------------------------------------------
__global__ __launch_bounds__(32) void k_tok(
    const float* __restrict__ lifted, const float* __restrict__ Wp,
    const float* __restrict__ bp, float* __restrict__ tok)
{
    const int m0 = blockIdx.x * 16, n0 = blockIdx.y * 16;
    const int lane = threadIdx.x & 31, l15 = lane & 15, hi = lane >> 4;
    v8f acc = {};
    const float* lr = lifted + (m0 + l15) * KD;
    for (int k = 0; k < KD; k += 4) {
        const int ka = k + 2 * hi;
        v2f a;  a.x = lr[ka];  a.y = lr[ka + 1];
        v2f bb;
        bb.x = Wp[ka       * DM + n0 + l15];
        bb.y = Wp[(ka + 1) * DM + n0 + l15];
        acc = wmma_f32x4(a, bb, acc);
    }
    const float bpv = bp[n0 + l15];
    #pragma unroll
    for (int r = 0; r < 8; ++r) {
        int m = m0 + r + 8 * hi;
        tok[m * DM + n0 + l15] = acc[r] + bpv;
    }
}

// ---------------------------------------------------------------------------
extern "C" void kernel_launch(void* const* d_in, const int* in_sizes, int n_in,
                              void* d_out, int out_size, void* d_ws, size_t ws_size,
                              hipStream_t stream)
{
    (void)in_sizes; (void)n_in; (void)out_size; (void)ws_size;

    const float* x  = (const float*)d_in[0];
    const float* W1 = (const float*)d_in[1];
    const float* b1 = (const float*)d_in[2];
    const float* W2 = (const float*)d_in[3];
    const float* b2 = (const float*)d_in[4];
    const float* Wl = (const float*)d_in[5];
    const float* bl = (const float*)d_in[6];
    const float* mu = (const float*)d_in[7];
    const float* sg = (const float*)d_in[8];
    const float* Wp = (const float*)d_in[9];
    const float* bp = (const float*)d_in[10];

    float* out = (float*)d_out;
    float* tok = out;                    // tokens: 32*16*1024
    float* ys  = out + BN * NPX * DM;    // y_star: 32*4096

    float* ws   = (float*)d_ws;
    float* sal  = ws;                    // 131072 floats
    float* cumv = ws + 131072;           // 512
    float* zbuf = ws + 131584;           // 512*260
    float* lift = ws + 264704;           // 512*256

    k_sal  <<<dim3(1024),   dim3(256), 0, stream>>>(x, W1, b1, W2, b2, sal);
    k_ystar<<<dim3(512),    dim3(256), 0, stream>>>(ys);
    k_cum  <<<dim3(BN),     dim3(256), 0, stream>>>(sal, cumv);
    k_z    <<<dim3(BN*NPX), dim3(256), 0, stream>>>(x, sal, cumv, mu, sg, zbuf);
    k_lift <<<dim3(32, 16), dim3(32),  0, stream>>>(zbuf, Wl, bl, lift);
    k_tok  <<<dim3(32, 64), dim3(32),  0, stream>>>(lift, Wp, bp, tok);
}